// CE_25872882991735
// MI455X (gfx1250) — compile-verified
//
#include <hip/hip_runtime.h>
#include <hip/hip_bf16.h>
#include <math.h>
#include <stdint.h>

// ---- problem constants (from reference) ----
#define NB   32
#define CC   256
#define HW   4096
#define GG   4
#define DD   64
#define MTOT (NB*HW)      // 131072 columns per group
#define EPSC 1e-5f
#define LNE  1e-5f

typedef __attribute__((ext_vector_type(2))) float v2f;
typedef __attribute__((ext_vector_type(8))) float v8f;

// LDS tile row stride (floats). 68 = 64 + 4 pad: column accesses hit distinct
// banks, and 68*4B keeps 16B alignment for b128 async copies.
#define TS 68

// ---- CDNA5 async global->LDS helpers (ASYNCcnt path) ----
// LDS byte offset = low 32 bits of a generic pointer to a __shared__ object
// (ISA 10.2: flat->LDS mapping truncates to addr[31:0]).
__device__ __forceinline__ uint32_t lds_off(const void* p) {
  return (uint32_t)(size_t)p;
}
__device__ __forceinline__ void async_copy_b128(uint32_t lds_byte, const void* gaddr) {
  asm volatile("global_load_async_to_lds_b128 %0, %1, off"
               :: "v"(lds_byte), "v"(gaddr) : "memory");
}
__device__ __forceinline__ void wait_async0() {
  asm volatile("s_wait_asynccnt 0x0" ::: "memory");
}

// =====================================================================
// Kernel 1: fused per-(n,c) sum/sumsq + per-group 64x64 Gram via WMMA.
// grid = G * N * 4 quarters, block = 128 (4 waves). Double-buffered
// async global->LDS staging overlapped with WMMA on the other buffer.
// =====================================================================
__global__ __launch_bounds__(128) void gram_stats_kernel(
    const float* __restrict__ X,
    float* __restrict__ sum_nc,    // [N*C]
    float* __restrict__ sumsq_nc,  // [N*C]
    float* __restrict__ gram)      // [G*64*64]
{
  const int SCH  = 4;
  const int wg   = blockIdx.x;
  const int g    = wg / (NB * SCH);
  const int rem  = wg % (NB * SCH);
  const int n    = rem / SCH;
  const int q    = rem % SCH;
  const int sbase = q * (HW / SCH);          // 1024-wide quarter

  const int t    = threadIdx.x;
  const int lane = t & 31;
  const int wv   = t >> 5;                   // wave id: owns d-tile row wv
  const int dld  = t >> 1;                   // channel this thread stages
  const int half = t & 1;                    // which 32-float half of the row

  __shared__ float tile[2][64 * TS];
  __shared__ float red[128];

  const float* xrow = X + ((size_t)(n * CC + g * DD + dld)) * HW + sbase + half * 32;
  const uint32_t lrow[2] = { lds_off(&tile[0][dld * TS + half * 32]),
                             lds_off(&tile[1][dld * TS + half * 32]) };

  v8f zero = {};
  v8f acc[4];
  #pragma unroll
  for (int e = 0; e < 4; ++e) acc[e] = zero;

  float lsum = 0.f, lss = 0.f;

  const int cl   = lane & 15;
  const int koff = (lane >> 4) * 2;

  // prologue: async-stage tile 0 into buffer 0
  #pragma unroll
  for (int j = 0; j < 8; ++j)
    async_copy_b128(lrow[0] + j * 16u, xrow + j * 4);
  wait_async0();
  __syncthreads();

  for (int it = 0; it < 16; ++it) {
    const int cur = it & 1;

    // issue async copies for the next tile into the other buffer
    if (it + 1 < 16) {
      const float* gsrc = xrow + (it + 1) * 64;
      #pragma unroll
      for (int j = 0; j < 8; ++j)
        async_copy_b128(lrow[cur ^ 1] + j * 16u, gsrc + j * 4);
    }

    // ---- per-channel stats from the staged tile ----
    {
      const float4* mine = (const float4*)(&tile[cur][dld * TS + half * 32]);
      #pragma unroll
      for (int j = 0; j < 8; ++j) {
        float4 v = mine[j];
        lsum += v.x + v.y + v.z + v.w;
        lss  += v.x * v.x + v.y * v.y + v.z * v.z + v.w * v.w;
      }
    }

    // ---- Gram tiles: wave wv = d-tile, loop e-tiles; K = 64 in steps of 4 ----
    const float* tl = tile[cur];
    const int rowA = wv * 16 + cl;
    #pragma unroll
    for (int k = 0; k < 64; k += 4) {
      v2f a;
      a.x = tl[rowA * TS + k + koff];
      a.y = tl[rowA * TS + k + koff + 1];
      #pragma unroll
      for (int e = 0; e < 4; ++e) {
        const int rowB = e * 16 + cl;
        v2f b;
        b.x = tl[rowB * TS + k + koff];
        b.y = tl[rowB * TS + k + koff + 1];
        acc[e] = __builtin_amdgcn_wmma_f32_16x16x4_f32(
            false, a, false, b, (short)0, acc[e], false, false);
      }
    }

    wait_async0();     // next buffer fully landed in LDS
    __syncthreads();   // everyone done reading current buffer
  }

  // ---- accumulate Gram into global (f32 atomics) ----
  float* gbase = gram + g * DD * DD;
  #pragma unroll
  for (int e = 0; e < 4; ++e) {
    #pragma unroll
    for (int r = 0; r < 8; ++r) {
      const int row = wv * 16 + r + ((lane >> 4) << 3);
      const int col = e * 16 + cl;
      atomicAdd(&gbase[row * DD + col], acc[e][r]);
    }
  }

  // ---- per-channel partial stats ----
  red[t] = lsum;
  __syncthreads();
  if (half == 0) atomicAdd(&sum_nc[n * CC + g * DD + dld], red[t] + red[t + 1]);
  __syncthreads();
  red[t] = lss;
  __syncthreads();
  if (half == 0) atomicAdd(&sumsq_nc[n * CC + g * DD + dld], red[t] + red[t + 1]);
}

// =====================================================================
// Kernel 2: tiny finishing work. block 0 = SE/MLP path, blocks 1..4 =
// per-group Sigma + Newton-Schulz (3 iters) on 64x64 in LDS.
// =====================================================================
__device__ inline void matmul64(float* Cm, const float* A, const float* B, int t) {
  // 256 threads: row r = t>>2, 16 cols at (t&3)*16
  const int r  = t >> 2;
  const int c0 = (t & 3) * 16;
  float acc[16];
  #pragma unroll
  for (int i = 0; i < 16; ++i) acc[i] = 0.f;
  for (int k = 0; k < 64; ++k) {
    const float a = A[r * 64 + k];
    const float* br = &B[k * 64 + c0];
    #pragma unroll
    for (int i = 0; i < 16; ++i) acc[i] += a * br[i];
  }
  #pragma unroll
  for (int i = 0; i < 16; ++i) Cm[r * 64 + c0 + i] = acc[i];
}

__global__ __launch_bounds__(256) void mid_kernel(
    const float* __restrict__ sum_nc, const float* __restrict__ sumsq_nc,
    const float* __restrict__ gram,
    const float* __restrict__ fc1, const float* __restrict__ lng,
    const float* __restrict__ lnb, const float* __restrict__ fc2,
    const float* __restrict__ xw,
    float* __restrict__ Pout, float* __restrict__ yscale,
    float* __restrict__ wblend)
{
  __shared__ float smem[16384];   // 64 KB carved per role below
  const int t = threadIdx.x;

  if (blockIdx.x == 0) {
    // ---------------- SE path ----------------
    float* xv   = smem;                   // 8192: x_var[n,c]
    float* hbuf = smem + 8192;            // 2048: h[n,j]
    float* redu = smem + 8192 + 2048;     // 256 : reduction

    float part = 0.f;
    for (int idx = t; idx < NB * CC; idx += 256) {
      const float s  = sum_nc[idx];
      const float ss = sumsq_nc[idx];
      const float var = (ss - s * s * (1.0f / HW)) * (1.0f / (HW - 1)); // ddof=1
      xv[idx] = var;
      part += var;
    }
    redu[t] = part;
    __syncthreads();
    for (int off = 128; off > 0; off >>= 1) {
      if (t < off) redu[t] += redu[t + off];
      __syncthreads();
    }
    const float scale = sqrtf(redu[0] * (1.0f / (NB * CC)));
    __syncthreads();

    // h = x_var @ fc1^T : [32 x 64]
    for (int idx = t; idx < NB * 64; idx += 256) {
      const int n = idx >> 6, j = idx & 63;
      const float* xr = &xv[n * CC];
      const float* wr = &fc1[j * CC];
      float a = 0.f;
      for (int c = 0; c < CC; ++c) a += xr[c] * wr[c];
      hbuf[idx] = a;
    }
    __syncthreads();

    // LayerNorm (biased var) + affine + ReLU, per row
    if (t < NB) {
      float mu = 0.f;
      for (int j = 0; j < 64; ++j) mu += hbuf[t * 64 + j];
      mu *= (1.0f / 64.0f);
      float vv = 0.f;
      for (int j = 0; j < 64; ++j) { const float d = hbuf[t * 64 + j] - mu; vv += d * d; }
      vv *= (1.0f / 64.0f);
      const float inv = rsqrtf(vv + LNE);
      for (int j = 0; j < 64; ++j) {
        float hh = (hbuf[t * 64 + j] - mu) * inv * lng[j] + lnb[j];
        hbuf[t * 64 + j] = hh > 0.f ? hh : 0.f;
      }
    }
    __syncthreads();

    // y = sigmoid(h @ fc2^T); gate = y / scale
    for (int idx = t; idx < NB * CC; idx += 256) {
      const int n = idx >> 8, c = idx & 255;
      const float* hr = &hbuf[n * 64];
      const float* wr = &fc2[c * 64];
      float a = 0.f;
      for (int j = 0; j < 64; ++j) a += hr[j] * wr[j];
      const float y = 1.0f / (1.0f + expf(-a));
      yscale[idx] = y / scale;
    }
    if (t == 0) wblend[0] = 1.0f / (1.0f + expf(-xw[0]));
  } else {
    // ---------------- group whitening P ----------------
    const int g = blockIdx.x - 1;
    float* S  = smem;             // Sigma_N
    float* P  = smem + 4096;
    float* T1 = smem + 8192;
    float* T2 = smem + 12288;
    __shared__ float chm[64];
    __shared__ float rtr;

    if (t < 64) {
      float s = 0.f;
      for (int n = 0; n < NB; ++n) s += sum_nc[n * CC + g * DD + t];
      chm[t] = s * (1.0f / MTOT);
    }
    __syncthreads();

    const float* gb = &gram[g * DD * DD];
    for (int idx = t; idx < DD * DD; idx += 256) {
      const int d = idx >> 6, e = idx & 63;
      const float cov = gb[idx] - (float)MTOT * chm[d] * chm[e];  // sum xc*xc^T
      S[idx] = EPSC * cov + (d == e ? (1.0f / MTOT) : 0.f);
      P[idx] = (d == e) ? 1.f : 0.f;
    }
    __syncthreads();
    if (t == 0) {
      float s = 0.f;
      for (int d = 0; d < DD; ++d) s += S[d * DD + d];
      rtr = 1.0f / s;
    }
    __syncthreads();
    for (int idx = t; idx < DD * DD; idx += 256) S[idx] *= rtr;
    __syncthreads();

    for (int itr = 0; itr < 3; ++itr) {
      matmul64(T1, P, P, t);  __syncthreads();
      matmul64(T2, T1, P, t); __syncthreads();
      matmul64(T1, T2, S, t); __syncthreads();
      for (int idx = t; idx < DD * DD; idx += 256)
        P[idx] = -0.5f * P[idx] + 1.5f * T1[idx];
      __syncthreads();
    }
    for (int idx = t; idx < DD * DD; idx += 256)
      Pout[g * DD * DD + idx] = P[idx];
  }
}

// =====================================================================
// Kernel 3: Xn = P @ x via WMMA, fused blend + store (2nd pass over X).
// grid = G * N * 4, block = 128. Double-buffered async X tiles; P[g]
// async-staged once. Wave wv owns s-tile wv, loops d-tiles.
// =====================================================================
__global__ __launch_bounds__(128) void out_kernel(
    const float* __restrict__ X,
    const float* __restrict__ Pg,      // [G*64*64]
    const float* __restrict__ yscale,  // [N*C]
    const float* __restrict__ wblend,  // [1]
    float* __restrict__ out)
{
  const int SCH  = 4;
  const int wg   = blockIdx.x;
  const int g    = wg / (NB * SCH);
  const int rem  = wg % (NB * SCH);
  const int n    = rem / SCH;
  const int q    = rem % SCH;
  const int sbase = q * (HW / SCH);

  const int t    = threadIdx.x;
  const int lane = t & 31;
  const int wv   = t >> 5;
  const int dld  = t >> 1;
  const int half = t & 1;

  __shared__ float Pm[64 * TS];
  __shared__ float Xt[2][64 * TS];
  __shared__ float yr[64];
  __shared__ float wsh;

  const float* xrow = X + ((size_t)(n * CC + g * DD + dld)) * HW + sbase + half * 32;
  const uint32_t lprow   = lds_off(&Pm[dld * TS + half * 32]);
  const uint32_t lxrow[2] = { lds_off(&Xt[0][dld * TS + half * 32]),
                              lds_off(&Xt[1][dld * TS + half * 32]) };

  // prologue: async-stage P[g] and X tile 0
  {
    const float* psrc = Pg + g * DD * DD + dld * 64 + half * 32;
    #pragma unroll
    for (int j = 0; j < 8; ++j) async_copy_b128(lprow + j * 16u, psrc + j * 4);
    #pragma unroll
    for (int j = 0; j < 8; ++j) async_copy_b128(lxrow[0] + j * 16u, xrow + j * 4);
  }
  if (t < 64) yr[t] = yscale[n * CC + g * DD + t];
  if (t == 0) wsh = wblend[0];
  wait_async0();
  __syncthreads();
  const float w   = wsh;
  const float omw = 1.0f - w;

  const int cl   = lane & 15;
  const int koff = (lane >> 4) * 2;
  const int scol = wv * 16 + cl;       // this wave's spatial column in the tile

  for (int it = 0; it < 16; ++it) {
    const int cur = it & 1;
    if (it + 1 < 16) {
      const float* gsrc = xrow + (it + 1) * 64;
      #pragma unroll
      for (int j = 0; j < 8; ++j)
        async_copy_b128(lxrow[cur ^ 1] + j * 16u, gsrc + j * 4);
    }

    const float* xt = Xt[cur];
    #pragma unroll
    for (int dt = 0; dt < 4; ++dt) {
      v8f acc = {};
      const int rowA = dt * 16 + cl;
      #pragma unroll
      for (int k = 0; k < 64; k += 4) {
        v2f a, b;
        a.x = Pm[rowA * TS + k + koff];            // A = P rows, K = channel e
        a.y = Pm[rowA * TS + k + koff + 1];
        b.x = xt[(k + koff)     * TS + scol];      // B = X tile, K = channel, N = s
        b.y = xt[(k + koff + 1) * TS + scol];
        acc = __builtin_amdgcn_wmma_f32_16x16x4_f32(
            false, a, false, b, (short)0, acc, false, false);
      }
      // fused blend + store: out = w*Xn + (1-w)*gate*X
      #pragma unroll
      for (int r = 0; r < 8; ++r) {
        const int d    = dt * 16 + r + ((lane >> 4) << 3);
        const float xv = xt[d * TS + scol];
        const float o  = w * acc[r] + omw * yr[d] * xv;
        out[((size_t)(n * CC + g * DD + d)) * HW + sbase + it * 64 + scol] = o;
      }
    }

    wait_async0();
    __syncthreads();
  }
}

// =====================================================================
extern "C" void kernel_launch(void* const* d_in, const int* in_sizes, int n_in,
                              void* d_out, int out_size, void* d_ws, size_t ws_size,
                              hipStream_t stream) {
  const float* X   = (const float*)d_in[0];
  const float* fc1 = (const float*)d_in[1];
  const float* lng = (const float*)d_in[2];
  const float* lnb = (const float*)d_in[3];
  const float* fc2 = (const float*)d_in[4];
  const float* xw  = (const float*)d_in[5];
  float* out = (float*)d_out;
  float* ws  = (float*)d_ws;

  float* sum_nc   = ws;             // 8192
  float* sumsq_nc = ws + 8192;      // 8192
  float* gram     = ws + 16384;     // 16384
  float* Pmat     = ws + 32768;     // 16384
  float* ysc      = ws + 49152;     // 8192
  float* wbl      = ws + 57344;     // 1

  // zero the accumulators (graph-capture safe)
  hipMemsetAsync(ws, 0, 32768 * sizeof(float), stream);

  gram_stats_kernel<<<GG * NB * 4, 128, 0, stream>>>(X, sum_nc, sumsq_nc, gram);
  mid_kernel<<<5, 256, 0, stream>>>(sum_nc, sumsq_nc, gram,
                                    fc1, lng, lnb, fc2, xw,
                                    Pmat, ysc, wbl);
  out_kernel<<<GG * NB * 4, 128, 0, stream>>>(X, Pmat, ysc, wbl, out);
}